// SelfStrAE_43396349558930
// MI455X (gfx1250) — compile-verified
//
#include <hip/hip_runtime.h>

// SelfStrAE greedy tree builder for MI455X (gfx1250, wave32).
// One workgroup (256 threads = 8 waves) per batch row; seq fully LDS-resident
// (~275 KB of the 320 KB/WGP); linked-list merge (no O(S*E) compaction);
// compose GEMM via V_WMMA_F32_16X16X4_F32 with register-resident B (W) and a
// zero-padded LDS staging buffer for branch-free A fetches.

typedef __attribute__((ext_vector_type(2))) float v2f;
typedef __attribute__((ext_vector_type(8))) float v8f;

#define SB   256          // batch
#define SS   128          // seq length
#define SE   512          // embedding dim
#define SM   (2*SS - 1)   // 255 tree nodes
#define EPSV 1e-8f
#define NEG_INF (-__builtin_inff())

__device__ __forceinline__ float waveSum(float v) {
#pragma unroll
  for (int off = 16; off > 0; off >>= 1) v += __shfl_xor(v, off, 32);
  return v;
}

__global__ void __launch_bounds__(256)
selfstrae_kernel(const int* __restrict__ tokens, const float* __restrict__ emb,
                 const float* __restrict__ W, const float* __restrict__ bias,
                 float* __restrict__ em, float* __restrict__ ad)
{
  extern __shared__ float smem[];
  float* seq     = smem;                 // SS*SE          (262144 B)
  float* hbuf    = seq + SS*SE;          // 16*128 A-stage (  8192 B)
  float* parentB = hbuf + 16*128;        // SE             (  2048 B)
  float* norms   = parentB + SE;         // SS
  float* cosv    = norms + SS;           // SS
  float* red     = cosv + SS;            // 8
  int*   nxt     = (int*)(red + 8);      // SS
  int*   prv     = nxt + SS;             // SS
  int*   trn     = prv + SS;             // SS
  int*   ctrl    = trn + SS;             // 2

  const int bb   = blockIdx.x;
  const int tid  = threadIdx.x;
  const int lane = tid & 31;
  const int wv   = tid >> 5;

  // ---- hoist B fragments (W tile for this wave) into registers ----
  // B[k][n] = W[n][k]; lanes 0-15: N=lane,   K = 4i+0 / 4i+1
  //                    lanes 16-31: N=lane-16, K = 4i+2 / 4i+3
  const int cL = lane & 15;
  const int kh = (lane >> 4) << 1;       // 0 or 2
  v2f breg[32];
  {
    const float* Wrow = W + (size_t)((wv & 3)*16 + cL)*128 + kh;
#pragma unroll
    for (int i = 0; i < 32; ++i)
      breg[i] = *(const v2f*)(Wrow + i*4);
  }
  const float bg = bias[(wv & 3)*16 + cL];

  // ---- Phase 0: gather leaf embeddings, init bookkeeping ----
  const size_t emBase = (size_t)bb * SM * SE;
  for (int s = wv; s < SS; s += 8) {
    const int tok = tokens[bb*SS + s];
    const float* src = emb + (size_t)tok * SE;
    for (int j = lane; j < SE; j += 32) {
      float v = src[j];
      seq[s*SE + j] = v;
      em[emBase + (size_t)s*SE + j] = v;   // leaf nodes 0..S-1
    }
  }
  for (int s = tid; s < SS; s += 256) {
    nxt[s] = (s + 1 < SS) ? s + 1 : -1;
    prv[s] = s - 1;
    trn[s] = s;
  }
  for (int i = tid; i < 8*128; i += 256) hbuf[8*128 + i] = 0.f;  // pad rows 8-15
  float* adrow = ad + (size_t)bb * SM * SM;
  for (int i = tid; i < SM*SM; i += 256) adrow[i] = 0.f;
  __syncthreads();

  // ---- norms of all leaves ----
  for (int s = wv; s < SS; s += 8) {
    float a = 0.f;
    for (int j = lane; j < SE; j += 32) { float v = seq[s*SE + j]; a += v*v; }
    a = waveSum(a);
    if (lane == 0) norms[s] = fmaxf(__builtin_sqrtf(a), EPSV);
  }
  __syncthreads();

  // ---- initial adjacent cosine sims ----
  for (int s = wv; s < SS; s += 8) {
    if (s < SS - 1) {
      float a = 0.f;
      for (int j = lane; j < SE; j += 32) a += seq[s*SE + j] * seq[(s+1)*SE + j];
      a = waveSum(a);
      if (lane == 0) cosv[s] = a / (norms[s] * norms[s+1]);
    } else if (lane == 0) {
      cosv[s] = NEG_INF;
    }
  }
  __syncthreads();

  // ---- 127 sequential merge steps ----
  for (int t = 0; t < SS - 1; ++t) {
    // argmax over cos (first-index tiebreak, matching jnp.argmax)
    if (wv == 0) {
      float best = NEG_INF; int bi = SS;
      for (int s = lane; s < SS; s += 32) {
        float c = cosv[s];
        if (c > best) { best = c; bi = s; }   // increasing s => keeps smallest
      }
#pragma unroll
      for (int off = 16; off > 0; off >>= 1) {
        float ob = __shfl_xor(best, off, 32);
        int   oi = __shfl_xor(bi,   off, 32);
        if (ob > best || (ob == best && oi < bi)) { best = ob; bi = oi; }
      }
      if (lane == 0) { ctrl[0] = bi; ctrl[1] = nxt[bi]; }
    }
    __syncthreads();
    const int m = ctrl[0];
    const int r = ctrl[1];
    const float* L = seq + m*SE;
    const float* R = seq + r*SE;
    const int dst = SS + t;
    float* emdst = em + emBase + (size_t)dst*SE;

    // ---- stage h = [left_c | right_c] into hbuf (rows 8-15 stay zero) ----
    for (int i = tid; i < 512; i += 256) {           // left halves
      const int c = i >> 6, f = i & 63;
      hbuf[c*128 + f] = L[i];
    }
    for (int i = tid; i < 512; i += 256) {           // right halves
      const int c = i >> 6, f = i & 63;
      hbuf[c*128 + 64 + f] = R[i];
    }
    __syncthreads();

    // ---- compose: out[c,g] = sum_f h[c,f]*W[g,f] ----
    // Waves 0..3 each own one 16-column N tile; K=128 as 32 x
    // V_WMMA_F32_16X16X4_F32, two accumulator chains, branch-free operands.
    if (wv < 4) {
      v8f acc0 = {}; v8f acc1 = {};
      const float* arow = hbuf + cL*128 + kh;        // A: M=cL, K base kh
#pragma unroll
      for (int kk = 0; kk < 32; kk += 2) {
        v2f a0 = *(const v2f*)(arow + kk*4);
        acc0 = __builtin_amdgcn_wmma_f32_16x16x4_f32(false, a0, false, breg[kk],
                                                     (short)0, acc0, false, false);
        v2f a1 = *(const v2f*)(arow + kk*4 + 4);
        acc1 = __builtin_amdgcn_wmma_f32_16x16x4_f32(false, a1, false, breg[kk+1],
                                                     (short)0, acc1, false, false);
      }
      // C/D layout: lanes 0-15 carry M=0..7 in VGPRs 0..7 (N = lane)
      if (lane < 16) {
        const int g = wv*16 + lane;
#pragma unroll
        for (int rr = 0; rr < 8; ++rr) {
          float v = acc0[rr] + acc1[rr] + bg;
          parentB[rr*64 + g] = v;           // stage (seq[r] unread hereafter)
          emdst[rr*64 + g]   = v;           // em[b, S+t, :]
        }
      }
    }
    __syncthreads();

    // ---- concurrently: new cosine dots (waves 0-2), commit parent (3-7) ----
    if (wv == 0) {                           // dot(seq[prev(m)], parent)
      const int p = prv[m];
      float a = 0.f;
      if (p >= 0) {
        const float* P = seq + p*SE;
        for (int j = lane; j < SE; j += 32) a += P[j] * parentB[j];
      }
      a = waveSum(a);
      if (lane == 0) red[0] = a;
    } else if (wv == 1) {                    // dot(parent, seq[next(r)])
      const int nn = nxt[r];
      float a = 0.f;
      if (nn >= 0) {
        const float* N = seq + nn*SE;
        for (int j = lane; j < SE; j += 32) a += N[j] * parentB[j];
      }
      a = waveSum(a);
      if (lane == 0) red[1] = a;
    } else if (wv == 2) {                    // ||parent||^2
      float a = 0.f;
      for (int j = lane; j < SE; j += 32) { float v = parentB[j]; a += v*v; }
      a = waveSum(a);
      if (lane == 0) red[2] = a;
    } else {                                 // waves 3-7: seq[r] = parent
      float* D = seq + r*SE;
      for (int j = (wv - 3)*32 + lane; j < SE; j += 160) D[j] = parentB[j];
    }
    __syncthreads();

    // ---- scalar bookkeeping (thread 0) ----
    if (tid == 0) {
      const float nr = fmaxf(__builtin_sqrtf(red[2]), EPSV);
      norms[r] = nr;
      const int p  = prv[m];
      const int nn = nxt[r];
      if (p >= 0) {
        nxt[p]  = r;
        cosv[p] = red[0] / (norms[p] * nr);
      }
      prv[r]  = p;
      cosv[r] = (nn >= 0) ? red[1] / (nr * norms[nn]) : NEG_INF;
      cosv[m] = NEG_INF;                    // slot m is dead
      adrow[dst*SM + trn[m]] = 1.f;
      adrow[dst*SM + trn[r]] = 1.f;
      trn[r] = dst;
    }
    __syncthreads();
  }
}

extern "C" void kernel_launch(void* const* d_in, const int* in_sizes, int n_in,
                              void* d_out, int out_size, void* d_ws, size_t ws_size,
                              hipStream_t stream) {
  const int*   tokens = (const int*)d_in[0];    // [B,S] int32
  const float* emb    = (const float*)d_in[1];  // [V,E]
  const float* W      = (const float*)d_in[2];  // [64,128]
  const float* bias   = (const float*)d_in[3];  // [64]
  float* em = (float*)d_out;                    // [B,M,E]
  float* ad = em + (size_t)SB * SM * SE;        // [B,M,M]

  // LDS: seq 262144 + hbuf 8192 + parent 2048 + norms/cos 1024 + red 32
  //      + nxt/prv/trn 1536 + ctrl 8  = 274984 bytes (< 320 KB/WGP)
  const size_t smemBytes =
      (size_t)(SS*SE + 16*128 + SE + SS + SS + 8) * sizeof(float) +
      (size_t)(3*SS + 2) * sizeof(int);

  selfstrae_kernel<<<SB, 256, smemBytes, stream>>>(tokens, emb, W, bias, em, ad);
}